// Memory_32753420599710
// MI455X (gfx1250) — compile-verified
//
#include <hip/hip_runtime.h>
#include <hip/hip_bf16.h>

#define KD      256
#define VD      512
#define NB      512
#define NSLOT   100000
#define RK      8
#define KT      9
#define NHEAD   8
#define HDIM    64
#define NEG_SLOPE 0.2f
#define EPSN    1e-8f

#define SLABS          25
#define SLAB_SZ        4000
#define TILES_PER_SLAB 250

// d_out float-element segment offsets (tuple concatenated flat, return order)
#define SEG_COMPOSED 0u            // [512][1024]
#define SEG_IDX      524288u       // [512][8] int32 (stored as raw bits)
#define SEG_SIM      528384u       // [512][8] f32
#define SEG_GATT     532480u       // [512][9][9][8] f32
#define SEG_SELIDX   864256u       // [512][8][1] f32

typedef __attribute__((ext_vector_type(2))) float v2f;
typedef __attribute__((ext_vector_type(8))) float v8f;

__device__ __forceinline__ v8f wmma_f32_16x16x4(v2f a, v2f b, v8f c) {
  // D = A(16x4,f32) * B(4x16,f32) + C(16x16,f32); wave32, one tile per wave
  return __builtin_amdgcn_wmma_f32_16x16x4_f32(false, a, false, b, (short)0, c,
                                               false, false);
}

// ---------------------------------------------------------------------------
// Kernel A: Y[m,:] = normalize(X[m,:] @ W^T + bias), rows per block = 16.
// X:[M,256], W:[256,256] (W[n][k]), Y:[M,256]. 256 threads = 8 waves, each
// wave computes two 16x16 output tiles via 64 chained v_wmma_f32_16x16x4_f32.
// ---------------------------------------------------------------------------
__global__ __launch_bounds__(256)
void proj_norm_kernel(const float* __restrict__ X, const float* __restrict__ W,
                      const float* __restrict__ bias, float* __restrict__ Y) {
  __shared__ float sA[16][KD + 4];
  __shared__ float sY[16][KD + 4];
  __shared__ float sPart[16][16];
  __shared__ float sInv[16];
  const int t  = threadIdx.x;
  const int m0 = blockIdx.x * 16;

  #pragma unroll 4
  for (int j = 0; j < 16; ++j)
    sA[j][t] = X[(size_t)(m0 + j) * KD + t];
  __syncthreads();

  const int lane = t & 31, w = t >> 5;
  const int l = lane & 15, hh = lane >> 4;
  for (int i = 0; i < 2; ++i) {
    const int n0 = (w * 2 + i) * 16;
    v8f acc = {};
    for (int k = 0; k < KD; k += 4) {
      v2f a, b;
      a.x = sA[l][k + 2 * hh];
      a.y = sA[l][k + 2 * hh + 1];
      const float* wr = W + (size_t)(n0 + l) * KD + k + 2 * hh;  // B[k][n]=W[n][k]
      b.x = wr[0];
      b.y = wr[1];
      acc = wmma_f32_16x16x4(a, b, acc);
    }
    const float bb = bias[n0 + l];
    #pragma unroll
    for (int r = 0; r < 8; ++r)
      sY[r + 8 * hh][n0 + l] = acc[r] + bb;
  }
  __syncthreads();

  { // per-row sum of squares
    const int r = t >> 4, c = t & 15;
    float p = 0.f;
    #pragma unroll
    for (int j = 0; j < 16; ++j) { float v = sY[r][c + 16 * j]; p += v * v; }
    sPart[r][c] = p;
  }
  __syncthreads();
  if (t < 16) {
    float s = 0.f;
    #pragma unroll
    for (int j = 0; j < 16; ++j) s += sPart[t][j];
    sInv[t] = 1.0f / fmaxf(sqrtf(s), EPSN);
  }
  __syncthreads();
  #pragma unroll 4
  for (int j = 0; j < 16; ++j)
    Y[(size_t)(m0 + j) * KD + t] = sY[j][t] * sInv[j];
}

// ---------------------------------------------------------------------------
// Kernel B: sim = PK @ PS^T fused with streaming per-row top-8.
// grid = (25 slabs, 32 row-blocks); 128 threads = 4 waves; each wave streams
// 16x16 sim tiles (WMMA), bounces them through LDS, lanes 0..15 keep an
// insertion-sorted top-8 per row in registers. Block-level merge -> partials.
// ---------------------------------------------------------------------------
__global__ __launch_bounds__(128)
void sim_topk_kernel(const float* __restrict__ PK, const float* __restrict__ PS,
                     float* __restrict__ pvals, int* __restrict__ pidx) {
  __shared__ float sP[16][KD + 4];
  __shared__ float sTile[4][16][17];
  __shared__ float sCV[4][16][RK];
  __shared__ int   sCI[4][16][RK];
  const int t    = threadIdx.x;
  const int slab = blockIdx.x;
  const int m0   = blockIdx.y * 16;

  #pragma unroll 4
  for (int j = 0; j < 32; ++j) {
    const int e = j * 128 + t;
    sP[e >> 8][e & 255] = PK[(size_t)(m0 + (e >> 8)) * KD + (e & 255)];
  }
  __syncthreads();

  const int lane = t & 31, w = t >> 5;
  const int l = lane & 15, hh = lane >> 4;
  float tv[RK]; int ti[RK];
  #pragma unroll
  for (int k = 0; k < RK; ++k) { tv[k] = -3.0e38f; ti[k] = 0; }

  const int base = slab * SLAB_SZ;
  for (int tt = w; tt < TILES_PER_SLAB; tt += 4) {   // wave-uniform loop
    const int s0 = base + tt * 16;
    v8f acc = {};
    for (int k = 0; k < KD; k += 4) {
      v2f a, b;
      a.x = sP[l][k + 2 * hh];
      a.y = sP[l][k + 2 * hh + 1];
      const float* pr = PS + (size_t)(s0 + l) * KD + k + 2 * hh; // B[k][n]=PS[n][k]
      b.x = pr[0];
      b.y = pr[1];
      acc = wmma_f32_16x16x4(a, b, acc);
    }
    #pragma unroll
    for (int r = 0; r < 8; ++r)
      sTile[w][r + 8 * hh][l] = acc[r];
    // same-wave LDS ops are in-order; lanes 0..15 do top-8 insertion
    if (lane < 16) {
      #pragma unroll
      for (int c = 0; c < 16; ++c) {
        const float v = sTile[w][lane][c];
        if (v > tv[RK - 1]) {
          tv[RK - 1] = v; ti[RK - 1] = s0 + c;
          #pragma unroll
          for (int q = RK - 1; q > 0; --q) {
            if (tv[q] > tv[q - 1]) {
              float fv = tv[q]; tv[q] = tv[q - 1]; tv[q - 1] = fv;
              int   iv = ti[q]; ti[q] = ti[q - 1]; ti[q - 1] = iv;
            }
          }
        }
      }
    }
  }
  if (lane < 16) {
    #pragma unroll
    for (int k = 0; k < RK; ++k) { sCV[w][lane][k] = tv[k]; sCI[w][lane][k] = ti[k]; }
  }
  __syncthreads();
  if (t < 16) {   // merge the 4 waves' lists for row t
    float mv[RK]; int mi[RK];
    #pragma unroll
    for (int k = 0; k < RK; ++k) { mv[k] = -3.0e38f; mi[k] = 0; }
    for (int ww = 0; ww < 4; ++ww)
      for (int k = 0; k < RK; ++k) {
        const float v = sCV[ww][t][k]; const int id = sCI[ww][t][k];
        if (v > mv[RK - 1]) {
          mv[RK - 1] = v; mi[RK - 1] = id;
          #pragma unroll
          for (int q = RK - 1; q > 0; --q) {
            if (mv[q] > mv[q - 1]) {
              float fv = mv[q]; mv[q] = mv[q - 1]; mv[q - 1] = fv;
              int   iv = mi[q]; mi[q] = mi[q - 1]; mi[q - 1] = iv;
            }
          }
        }
      }
    const size_t o = ((size_t)(m0 + t) * SLABS + slab) * RK;
    #pragma unroll
    for (int k = 0; k < RK; ++k) { pvals[o + k] = mv[k]; pidx[o + k] = mi[k]; }
  }
}

// ---------------------------------------------------------------------------
// Kernel C: final top-8 merge (200 candidates/row) + gathered outputs.
// ---------------------------------------------------------------------------
__global__ __launch_bounds__(256)
void merge_topk_kernel(float* __restrict__ pvals, const int* __restrict__ pidx,
                       const float* __restrict__ idx_store,
                       int* __restrict__ topidx_ws, float* __restrict__ dout) {
  const int row = blockIdx.x * 256 + threadIdx.x;
  if (row >= NB) return;
  const size_t base = (size_t)row * SLABS * RK;
  int*   out_idx  = (int*)(dout + SEG_IDX);
  float* out_sim  = dout + SEG_SIM;
  float* out_sidx = dout + SEG_SELIDX;
  for (int k = 0; k < RK; ++k) {
    float best = -3.0e38f; int bp = 0;
    for (int c = 0; c < SLABS * RK; ++c) {
      const float v = pvals[base + c];
      if (v > best) { best = v; bp = c; }
    }
    const int id = pidx[base + bp];
    pvals[base + bp] = -3.0e38f;          // rewritten by kernel B each launch
    out_idx[row * RK + k]  = id;
    out_sim[row * RK + k]  = best;
    out_sidx[row * RK + k] = idx_store[id];
    topidx_ws[row * RK + k] = id;
  }
}

// ---------------------------------------------------------------------------
// Kernel D: GAT compose + pooling + count-sketch + circular conv, one batch
// row per block (256 threads = 8 waves). Wh GEMM = [16x512]@[512x512] WMMA.
// ---------------------------------------------------------------------------
__global__ __launch_bounds__(256)
void compose_kernel(const float* __restrict__ outin,
                    const float* __restrict__ vstore,
                    const float* __restrict__ GW,      // gat_w [512][512] row-major
                    const float* __restrict__ a_src, const float* __restrict__ a_dst,
                    const float* __restrict__ att_w, const float* __restrict__ att_b,
                    const int* __restrict__ h1, const int* __restrict__ h2,
                    const float* __restrict__ s1, const float* __restrict__ s2,
                    const int* __restrict__ topidx, float* __restrict__ dout) {
  __shared__ float sF[16][VD + 8];          // feats, later reused as gat_out
  __shared__ float sWh[KT][VD + 8];
  __shared__ float sSrc[KT][NHEAD], sDst[KT][NHEAD];
  __shared__ float sAtt[KT][KT][NHEAD];
  __shared__ float sPsi1[VD], sPsi2[VD];
  __shared__ float sWk[KT];
  const int t = threadIdx.x;
  const int b = blockIdx.x;

  sPsi1[t] = 0.f; sPsi1[t + 256] = 0.f;
  sPsi2[t] = 0.f; sPsi2[t + 256] = 0.f;
  #pragma unroll 2
  for (int j = 0; j < 32; ++j) {
    const int e = j * 256 + t;
    const int row = e >> 9, col = e & 511;
    float v;
    if (row == 0)      v = outin[(size_t)b * VD + col];
    else if (row < KT) v = vstore[(size_t)topidx[b * RK + row - 1] * VD + col];
    else               v = 0.f;
    sF[row][col] = v;
  }
  __syncthreads();

  // psi1 = count_sketch(out_row)
  #pragma unroll
  for (int j = 0; j < 2; ++j) {
    const int d = t + 256 * j;
    atomicAdd(&sPsi1[h1[d]], sF[0][d] * s1[d]);
  }

  // Wh = feats(16x512) @ gat_w(512x512): 8 waves x 4 col tiles x 128 k-steps
  const int lane = t & 31, w = t >> 5;
  const int l = lane & 15, hh = lane >> 4;
  for (int i = 0; i < 4; ++i) {
    const int n0 = (w * 4 + i) * 16;
    v8f acc = {};
    for (int k = 0; k < VD; k += 4) {
      v2f a, bv;
      a.x = sF[l][k + 2 * hh];
      a.y = sF[l][k + 2 * hh + 1];
      const float* gr = GW + (size_t)(k + 2 * hh) * VD + n0 + l;  // B row-major
      bv.x = gr[0];
      bv.y = gr[VD];
      acc = wmma_f32_16x16x4(a, bv, acc);
    }
    #pragma unroll
    for (int r = 0; r < 8; ++r) {
      const int row = r + 8 * hh;
      if (row < KT) sWh[row][n0 + l] = acc[r];
    }
  }
  __syncthreads();

  if (t < KT * NHEAD) {               // src/dst projections
    const int k = t / NHEAD, h = t % NHEAD;
    float ss = 0.f, dd = 0.f;
    #pragma unroll 8
    for (int e = 0; e < HDIM; ++e) {
      const float wv = sWh[k][h * HDIM + e];
      ss += wv * a_src[h * HDIM + e];
      dd += wv * a_dst[h * HDIM + e];
    }
    sSrc[k][h] = ss; sDst[k][h] = dd;
  }
  __syncthreads();

  if (t < KT * NHEAD) {               // masked leaky-relu softmax attention
    const int ki = t / NHEAD, h = t % NHEAD;
    float lg[KT]; float mx = -3.0e38f;
    #pragma unroll
    for (int kj = 0; kj < KT; ++kj) {
      const bool conn = (ki == kj) || (ki == 0);
      const float x = sSrc[ki][h] + sDst[kj][h];
      const float lr = x > 0.f ? x : NEG_SLOPE * x;
      lg[kj] = conn ? lr : -1.0e9f;
      mx = fmaxf(mx, lg[kj]);
    }
    float sum = 0.f;
    #pragma unroll
    for (int kj = 0; kj < KT; ++kj) { lg[kj] = __expf(lg[kj] - mx); sum += lg[kj]; }
    const float inv = 1.0f / sum;
    #pragma unroll
    for (int kj = 0; kj < KT; ++kj) {
      const float a = lg[kj] * inv;
      sAtt[ki][kj][h] = a;
      dout[SEG_GATT + (((size_t)b * KT + ki) * KT + kj) * NHEAD + h] = a;
    }
  }
  __syncthreads();

  // gat_out = elu(att @ Wh), reusing sF rows 0..8
  #pragma unroll 2
  for (int j = 0; j < 18; ++j) {
    const int e = j * 256 + t;
    const int ki = e >> 9, n = e & 511, h = n >> 6;
    float s = 0.f;
    #pragma unroll
    for (int kj = 0; kj < KT; ++kj) s += sAtt[ki][kj][h] * sWh[kj][n];
    sF[ki][n] = s > 0.f ? s : (__expf(s) - 1.0f);
  }
  __syncthreads();

  if (t < KT) {                        // pooling logits
    float s = 0.f;
    for (int d = 0; d < VD; ++d) s += sF[t][d] * att_w[d];
    sWk[t] = s + att_b[0];
  }
  __syncthreads();
  if (t == 0) {                        // softmax over 9 nodes
    float mx = sWk[0];
    for (int k = 1; k < KT; ++k) mx = fmaxf(mx, sWk[k]);
    float sum = 0.f;
    for (int k = 0; k < KT; ++k) { sWk[k] = __expf(sWk[k] - mx); sum += sWk[k]; }
    const float inv = 1.f / sum;
    for (int k = 0; k < KT; ++k) sWk[k] *= inv;
  }
  __syncthreads();

  #pragma unroll
  for (int j = 0; j < 2; ++j) {        // pooled + psi2 count-sketch
    const int d = t + 256 * j;
    float p = 0.f;
    #pragma unroll
    for (int k = 0; k < KT; ++k) p += sWk[k] * sF[k][d];
    atomicAdd(&sPsi2[h2[d]], p * s2[d]);
  }
  __syncthreads();

  #pragma unroll
  for (int j = 0; j < 2; ++j) {        // circular conv == irfft(rfft*rfft)
    const int tt = t + 256 * j;
    float s = 0.f;
    for (int d = 0; d < VD; ++d)
      s += sPsi1[d] * sPsi2[(tt - d) & (VD - 1)];
    dout[(size_t)b * 2 * VD + tt]      = s;
    dout[(size_t)b * 2 * VD + VD + tt] = outin[(size_t)b * VD + tt];
  }
}

// ---------------------------------------------------------------------------
extern "C" void kernel_launch(void* const* d_in, const int* in_sizes, int n_in,
                              void* d_out, int out_size, void* d_ws, size_t ws_size,
                              hipStream_t stream) {
  (void)in_sizes; (void)n_in; (void)out_size; (void)ws_size;
  const float* outin  = (const float*)d_in[0];
  const float* qk     = (const float*)d_in[1];
  const float* kstore = (const float*)d_in[3];
  const float* vstore = (const float*)d_in[4];
  const float* istore = (const float*)d_in[5];
  const float* kp_w   = (const float*)d_in[6];
  const float* kp_b   = (const float*)d_in[7];
  const float* gw     = (const float*)d_in[8];
  const float* asrc   = (const float*)d_in[9];
  const float* adst   = (const float*)d_in[10];
  const float* attw   = (const float*)d_in[11];
  const float* attb   = (const float*)d_in[12];
  const int*   h1     = (const int*)d_in[13];
  const int*   h2     = (const int*)d_in[14];
  const float* s1     = (const float*)d_in[15];
  const float* s2     = (const float*)d_in[16];
  float* dout = (float*)d_out;
  float* ws   = (float*)d_ws;

  float* pk_norm = ws;                                   // 512*256
  float* ps_norm = pk_norm + (size_t)NB * KD;            // 100000*256
  float* pvals   = ps_norm + (size_t)NSLOT * KD;         // 512*25*8
  int*   pidx    = (int*)(pvals + (size_t)NB * SLABS * RK);
  int*   topidx  = pidx + (size_t)NB * SLABS * RK;       // 512*8

  proj_norm_kernel<<<NB / 16,    256, 0, stream>>>(qk,     kp_w, kp_b, pk_norm);
  proj_norm_kernel<<<NSLOT / 16, 256, 0, stream>>>(kstore, kp_w, kp_b, ps_norm);
  sim_topk_kernel<<<dim3(SLABS, NB / 16), 128, 0, stream>>>(pk_norm, ps_norm,
                                                            pvals, pidx);
  merge_topk_kernel<<<NB / 256, 256, 0, stream>>>(pvals, pidx, istore, topidx, dout);
  compose_kernel<<<NB, 256, 0, stream>>>(outin, vstore, gw, asrc, adst, attw, attb,
                                         h1, h2, s1, s2, topidx, dout);
}